// AlignNetwork_20564303413801
// MI455X (gfx1250) — compile-verified
//
#include <hip/hip_runtime.h>
#include <hip/hip_bf16.h>

typedef __attribute__((ext_vector_type(2))) float v2f;
typedef __attribute__((ext_vector_type(8))) float v8f;
typedef __attribute__((ext_vector_type(4))) float f4v;

#define NB    8
#define HH    128
#define WW    128
#define HW    (HH * WW)
#define C64   64
#define KKT   9          // 3x3 taps
#define KDIM  576        // 64 * 9

__device__ __forceinline__ v8f wmma_f32x4(v2f a, v2f b, v8f c) {
  // D(16x16,f32) = A(16x4,f32) * B(4x16,f32) + C
  return __builtin_amdgcn_wmma_f32_16x16x4_f32(
      /*neg_a=*/false, a, /*neg_b=*/false, b,
      /*c_mod=*/(short)0, c, /*reuse_a=*/false, /*reuse_b=*/false);
}

// Async copy 4 bytes global -> LDS (tracked by ASYNCcnt, no VGPR data path).
__device__ __forceinline__ void async_g2l_b32(const float* gptr, float* lptr) {
  const unsigned lds_off = (unsigned)(uintptr_t)lptr;  // low 32 bits = LDS offset
  asm volatile("global_load_async_to_lds_b32 %0, %1, off"
               :: "v"(lds_off), "v"(gptr)
               : "memory");
}

__device__ __forceinline__ void wait_asynccnt0() {
  asm volatile("s_wait_asynccnt 0x0" ::: "memory");
}

// ---------------------------------------------------------------------------
// Implicit-GEMM convolution, stride 1, pad KR/2.
//   M = 32 pixels along W per block, N = COUT_PAD (16 per wave, 2x(COUT_PAD/16)
//   waves), K = CIN*KR*KR stepped by 4 via V_WMMA_F32_16X16X4_F32.
//   Activations async-staged into LDS (halo zero-filled via DS); weights
//   streamed from L2 with prefetch.
// ---------------------------------------------------------------------------
template <int CIN, int C0, int KR, int COUT_PAD, int COUT_REAL, bool TWO_IN, bool LEAKY>
__global__ __launch_bounds__(64 * (COUT_PAD / 16))
void conv_wmma_kernel(const float* __restrict__ xA, const float* __restrict__ xB,
                      const float* __restrict__ w, const float* __restrict__ bias,
                      float* __restrict__ out) {
  constexpr int PAD   = KR / 2;
  constexpr int TW    = 32 + KR - 1;
  constexpr int NWAVE = 2 * (COUT_PAD / 16);
  constexpr int NTHR  = NWAVE * 32;
  constexpr int CSLAB = (CIN > 64) ? 64 : CIN;
  constexpr int NSLAB = CIN / CSLAB;
  constexpr int KSLAB = CSLAB * KR * KR;

  __shared__ float patch[CSLAB * KR * TW];

  const int tid = threadIdx.x;
  const int xt  = blockIdx.x % (WW / 32);
  const int y   = (blockIdx.x / (WW / 32)) % HH;
  const int b   = blockIdx.x / ((WW / 32) * HH);
  const int x0  = xt * 32;

  const int wave = tid >> 5;
  const int lane = tid & 31;
  const int wm   = wave & 1;        // M-tile: 0 or 16
  const int wn   = wave >> 1;       // N-tile
  const int hi   = lane >> 4;       // K half select
  const int ml   = lane & 15;       // A row / B col within tile
  const int m    = wm * 16 + ml;    // pixel row for A reads
  const int n    = wn * 16 + ml;    // output channel for B reads / D col

  const bool  nvalid = (n < COUT_REAL);
  const float* wrow  = w + (size_t)(nvalid ? n : 0) * (CIN * KR * KR);

  v8f acc = {0.f, 0.f, 0.f, 0.f, 0.f, 0.f, 0.f, 0.f};

  for (int s = 0; s < NSLAB; ++s) {
    __syncthreads();
    const int kgbase = s * KSLAB;
    // pull the weight slab toward the WGP while we stage activations
    __builtin_prefetch(wrow + kgbase, 0, 1);
    __builtin_prefetch(wrow + kgbase + KSLAB / 2, 0, 1);

    // ---- stage input patch for channels [s*CSLAB, s*CSLAB+CSLAB) ----
    // Interior: async global->LDS copy (ASYNCcnt).  Halo: DS zero store.
    for (int e = tid; e < CSLAB * KR * TW; e += NTHR) {
      const int rx = e % TW;
      const int t  = e / TW;
      const int ry = t % KR;
      const int cl = t / KR;
      const int c  = s * CSLAB + cl;
      const int gy = y - PAD + ry;
      const int gx = x0 - PAD + rx;
      if (gy >= 0 && gy < HH && gx >= 0 && gx < WW) {
        const float* src = (!TWO_IN || c < C0)
                               ? (xA + (size_t)(b * C0 + c) * HW)
                               : (xB + (size_t)(b * C0 + (c - C0)) * HW);
        async_g2l_b32(src + gy * WW + gx, &patch[e]);
      } else {
        patch[e] = 0.f;
      }
    }
    wait_asynccnt0();
    __syncthreads();

    // ---- K loop over this slab ----
    for (int kb = 0; kb < KSLAB; kb += 4) {
      const int k0  = kb + 2 * hi;
      const int k1  = k0 + 1;
      const int c0k = k0 / (KR * KR);
      const int r0  = k0 - c0k * (KR * KR);
      const int ky0 = r0 / KR;
      const int kx0 = r0 - ky0 * KR;
      const int c1k = k1 / (KR * KR);
      const int r1  = k1 - c1k * (KR * KR);
      const int ky1 = r1 / KR;
      const int kx1 = r1 - ky1 * KR;

      v2f a, bm;
      a.x  = patch[(c0k * KR + ky0) * TW + m + kx0];
      a.y  = patch[(c1k * KR + ky1) * TW + m + kx1];
      bm.x = wrow[kgbase + k0];
      bm.y = wrow[kgbase + k1];
      if (!nvalid) { bm.x = 0.f; bm.y = 0.f; }
      acc = wmma_f32x4(a, bm, acc);
    }
  }

  // ---- epilogue: bias (+leaky), store D ----
  if (nvalid) {
    const float  bv    = bias[n];
    const size_t obase = ((size_t)(b * COUT_REAL + n)) * HW + (size_t)y * WW + x0;
#pragma unroll
    for (int r = 0; r < 8; ++r) {
      const int mm = wm * 16 + r + 8 * hi;   // D row per ISA C/D layout
      float v = acc[r] + bv;
      if (LEAKY) v = (v >= 0.f) ? v : 0.1f * v;
      out[obase + mm] = v;
    }
  }
}

// ---------------------------------------------------------------------------
// Modulated deformable conv reduction:
//   per block: 16 pixels; build samp A-tile [16 x 576] in LDS from bilinear
//   gathers of Fe (4 precomputed corner weights*mask per (pixel, tap)), then
//   GEMM against w_dcn [64 x 576] with 144 f32 WMMA steps per wave.
// ---------------------------------------------------------------------------
__global__ __launch_bounds__(128)
void dcn_wmma_kernel(const float* __restrict__ fe, const float* __restrict__ oo,
                     const float* __restrict__ wd, const float* __restrict__ bd,
                     float* __restrict__ ff /* = d_out + B*18*HW (Ff base) */) {
  __shared__ float cw[16 * KKT][4];
  __shared__ int   cidx[16 * KKT][4];
  __shared__ float As[16 * 580];   // stride 580 to avoid bank conflicts

  const int tid = threadIdx.x;
  const int xt  = blockIdx.x % (WW / 16);
  const int y   = (blockIdx.x / (WW / 16)) % HH;
  const int b   = blockIdx.x / ((WW / 16) * HH);
  const int x0  = xt * 16;

  // ---- phase 1: sampling meta per (pixel m, tap kk) ----
  for (int e = tid; e < 16 * KKT; e += 128) {
    const int m  = e / KKT;
    const int kk = e - m * KKT;
    const int xg = x0 + m;
    const float* op = oo + (size_t)(b * 27) * HW + (size_t)y * WW + xg;
    const float o1   = op[(size_t)kk * HW];
    const float o2   = op[(size_t)(9 + kk) * HW];
    const float mlog = op[(size_t)(18 + kk) * HW];
    const float mask = 1.f / (1.f + __expf(-mlog));
    const int   ky = kk / 3, kx = kk - ky * 3;
    const float ys  = (float)(y - 1 + ky) + o1;
    const float xs  = (float)(xg - 1 + kx) + o2;
    const float y0f = floorf(ys), x0f = floorf(xs);
    const float fy = ys - y0f, fx = xs - x0f;
    const int   yi = (int)y0f, xi = (int)x0f;
#pragma unroll
    for (int j = 0; j < 4; ++j) {
      const int  dy = j >> 1, dx = j & 1;
      const int  yy = yi + dy, xx = xi + dx;
      const float wgt  = (dy ? fy : 1.f - fy) * (dx ? fx : 1.f - fx);
      const bool valid = (yy >= 0) && (yy < HH) && (xx >= 0) && (xx < WW);
      const int  yc = min(max(yy, 0), HH - 1);
      const int  xc = min(max(xx, 0), WW - 1);
      cw[e][j]   = valid ? wgt * mask : 0.f;
      cidx[e][j] = yc * WW + xc;
    }
  }
  __syncthreads();

  // ---- phase 2: materialize samp A-tile in LDS ----
  const float* feb = fe + (size_t)(b * C64) * HW;
  for (int e = tid; e < 16 * KDIM; e += 128) {
    const int m  = e / KDIM;
    const int k  = e - m * KDIM;
    const int c  = k / KKT;
    const int kk = k - c * KKT;
    const int mi = m * KKT + kk;
    const float* fc = feb + (size_t)c * HW;
    As[m * 580 + k] = cw[mi][0] * fc[cidx[mi][0]] + cw[mi][1] * fc[cidx[mi][1]] +
                      cw[mi][2] * fc[cidx[mi][2]] + cw[mi][3] * fc[cidx[mi][3]];
  }
  __syncthreads();

  // ---- phase 3: WMMA GEMM, 4 waves x 16 output channels ----
  const int wave = tid >> 5;
  const int lane = tid & 31;
  const int hi   = lane >> 4;
  const int ml   = lane & 15;
  const int n    = wave * 16 + ml;
  const float* wrow = wd + (size_t)n * KDIM;
  __builtin_prefetch(wrow, 0, 1);

  v8f acc = {0.f, 0.f, 0.f, 0.f, 0.f, 0.f, 0.f, 0.f};
  for (int kb = 0; kb < KDIM; kb += 4) {
    const int k0 = kb + 2 * hi;
    v2f a, bm;
    a.x  = As[ml * 580 + k0];
    a.y  = As[ml * 580 + k0 + 1];
    bm.x = wrow[k0];
    bm.y = wrow[k0 + 1];
    acc = wmma_f32x4(a, bm, acc);
  }

  const float  bv    = bd[n];
  const size_t obase = ((size_t)(b * 128) + 64 + n) * HW + (size_t)y * WW + x0;
#pragma unroll
  for (int r = 0; r < 8; ++r) {
    ff[obase + r + 8 * hi] = acc[r] + bv;
  }
}

// ---------------------------------------------------------------------------
// offset_out = oo channels [0,18); Ff[:, 0:64] = Fi   (float4 vectorized)
// ---------------------------------------------------------------------------
__global__ __launch_bounds__(256)
void offset_copy_kernel(const float* __restrict__ oo, float* __restrict__ out) {
  const size_t i = (size_t)blockIdx.x * 256 + threadIdx.x;       // in float4
  const size_t total = (size_t)NB * 18 * HW / 4;
  if (i >= total) return;
  const size_t per = (size_t)18 * HW / 4;
  const size_t b = i / per, p = i - b * per;
  ((f4v*)out)[i] = ((const f4v*)oo)[b * ((size_t)27 * HW / 4) + p];
}

__global__ __launch_bounds__(256)
void fi_copy_kernel(const float* __restrict__ fi, float* __restrict__ ff) {
  const size_t i = (size_t)blockIdx.x * 256 + threadIdx.x;       // in float4
  const size_t total = (size_t)NB * C64 * HW / 4;
  if (i >= total) return;
  const size_t per = (size_t)C64 * HW / 4;
  const size_t b = i / per, p = i - b * per;
  ((f4v*)ff)[b * ((size_t)128 * HW / 4) + p] = ((const f4v*)fi)[i];
}

// ---------------------------------------------------------------------------
extern "C" void kernel_launch(void* const* d_in, const int* in_sizes, int n_in,
                              void* d_out, int out_size, void* d_ws, size_t ws_size,
                              hipStream_t stream) {
  const float* Fi    = (const float*)d_in[0];
  const float* Fe    = (const float*)d_in[1];
  const float* w1    = (const float*)d_in[2];
  const float* b1    = (const float*)d_in[3];
  const float* w2    = (const float*)d_in[4];
  const float* b2    = (const float*)d_in[5];
  const float* w_off = (const float*)d_in[6];
  const float* b_off = (const float*)d_in[7];
  const float* w_dcn = (const float*)d_in[8];
  const float* b_dcn = (const float*)d_in[9];
  float* out = (float*)d_out;

  // workspace: feat1 | feat2 ; offset-conv output reuses feat1 (dead by then)
  float* feat1 = (float*)d_ws;                           // B*64*HW
  float* feat2 = feat1 + (size_t)NB * C64 * HW;          // B*64*HW
  float* oo    = feat1;                                  // B*27*HW (reuse)

  float* ff = out + (size_t)NB * 18 * HW;                // Ff base in d_out

  const int conv_grid = NB * HH * (WW / 32);             // 4096
  const int dcn_grid  = NB * HH * (WW / 16);             // 8192

  // conv1: concat(Fi,Fe) [128ch] -> feat1, 5x5 pad2, leaky
  conv_wmma_kernel<128, 64, 5, 64, 64, true, true>
      <<<conv_grid, 256, 0, stream>>>(Fi, Fe, w1, b1, feat1);

  // conv2: feat1 -> feat2, 3x3 pad1, leaky
  conv_wmma_kernel<64, 64, 3, 64, 64, false, true>
      <<<conv_grid, 256, 0, stream>>>(feat1, nullptr, w2, b2, feat2);

  // offset head: feat2 -> oo [27ch], 3x3 pad1, no activation
  conv_wmma_kernel<64, 64, 3, 32, 27, false, false>
      <<<conv_grid, 128, 0, stream>>>(feat2, nullptr, w_off, b_off, oo);

  // offset_out = oo[:, 0:18]
  {
    const size_t tot = (size_t)NB * 18 * HW / 4;
    offset_copy_kernel<<<(unsigned)((tot + 255) / 256), 256, 0, stream>>>(oo, out);
  }
  // Ff[:, 0:64] = Fi
  {
    const size_t tot = (size_t)NB * C64 * HW / 4;
    fi_copy_kernel<<<(unsigned)((tot + 255) / 256), 256, 0, stream>>>(Fi, ff);
  }
  // Ff[:, 64:128] = DCN(Fe, oo) with w_dcn, b_dcn
  dcn_wmma_kernel<<<dcn_grid, 128, 0, stream>>>(Fe, oo, w_dcn, b_dcn, ff);
}